// DilatedAttention_16320875724876
// MI455X (gfx1250) — compile-verified
//
#include <hip/hip_runtime.h>

typedef __attribute__((ext_vector_type(2))) float v2f;
typedef __attribute__((ext_vector_type(4))) float v4f;
typedef __attribute__((ext_vector_type(8))) float v8f;

#define D_MODEL 768
#define SEQ     16384
#define SEG     512
#define LSEG    256      // dilated tokens per segment
#define NSEG    32
#define BATCH   4
#define QT      16       // q rows per block tile
#define SROW    260      // padded LDS score-row stride: 260%64=4 -> conflict-free column reads
#define QROW    772      // padded LDS Q-row stride:     772%64=4 -> conflict-free column reads

__launch_bounds__(256)
__global__ void dilated_attn_kernel(const float* __restrict__ X, float* __restrict__ O)
{
    __shared__ float Qsh[QT * QROW];   // 16 x 772 fp32 staged Q tile (~48 KB)
    __shared__ float Ssh[QT * SROW];   // 16 x 260 fp32 scores -> exp(p) (unnormalized)
    __shared__ float red[16 * 16];
    __shared__ float rowmax[16];
    __shared__ float rinv[16];

    const int tid  = threadIdx.x;
    const int lane = tid & 31;
    const int w    = tid >> 5;     // wave 0..7
    const int hl   = lane >> 4;    // half-wave group 0/1
    const int ll   = lane & 15;

    const int qt = blockIdx.x & 15;            // q-tile within segment
    const int n  = (blockIdx.x >> 4) & 31;     // segment
    const int b  = blockIdx.x >> 9;            // batch

    // first sequence row of this segment; dilated token i lives at seq row 2*i
    const float* seg = X + ((size_t)b * SEQ + (size_t)n * SEG) * D_MODEL;

    // ---------------- Stage Q tile into LDS (coalesced b128 loads, once per block) -----
    {
        const int qrow = tid >> 4;                       // 16 threads per row
        const float* src = seg + (size_t)(2 * (qt * QT + qrow)) * D_MODEL;
        float* dst = &Qsh[qrow * QROW];
        #pragma unroll
        for (int i = 0; i < 12; ++i) {
            const int c4 = ((tid & 15) + 16 * i) * 4;    // float4 index within the row
            *(v4f*)(dst + c4) = *(const v4f*)(src + c4);
        }
    }
    __syncthreads();

    // ---------------- Pass 1: S = Q K^T  (f32 WMMA, K-dim = 768, step 4) ----------------
    // A-frag from LDS: lane ll -> Q row ll, k = k0 + 2*hl + {0,1} (ds_load_b64)
    // B-frags from global: two N-tiles per wave
    const float* lA    = &Qsh[ll * QROW + 2 * hl];
    const float* b0Ptr = seg + (size_t)(2 * ((2 * w + 0) * 16 + ll)) * D_MODEL + 2 * hl;
    const float* b1Ptr = seg + (size_t)(2 * ((2 * w + 1) * 16 + ll)) * D_MODEL + 2 * hl;

    v8f acc0 = {};
    v8f acc1 = {};
    for (int k0 = 0; k0 < D_MODEL; k0 += 4) {
        v2f a  = *(const v2f*)(lA + k0);
        v2f b0 = *(const v2f*)(b0Ptr + k0);
        v2f b1 = *(const v2f*)(b1Ptr + k0);
        acc0 = __builtin_amdgcn_wmma_f32_16x16x4_f32(false, a, false, b0, (short)0, acc0, false, false);
        acc1 = __builtin_amdgcn_wmma_f32_16x16x4_f32(false, a, false, b1, (short)0, acc1, false, false);
    }
    // D element: (M = v + 8*hl, N = nt*16 + ll)
    {
        const int c0 = (2 * w + 0) * 16 + ll;
        const int c1 = (2 * w + 1) * 16 + ll;
        #pragma unroll
        for (int v = 0; v < 8; ++v) {
            const int m = v + 8 * hl;
            Ssh[m * SROW + c0] = acc0[v];
            Ssh[m * SROW + c1] = acc1[v];
        }
    }
    __syncthreads();

    // ---------------- Softmax (scale folded into exp; 1/sum folded into final store) ----
    const float scale = 0.03608439182435161f;  // 1/sqrt(768)
    {
        const int row = tid & 15, chunk = tid >> 4;
        float* rowp = &Ssh[row * SROW + chunk * 16];
        float mx = -3.4e38f;
        #pragma unroll
        for (int c = 0; c < 16; ++c) mx = fmaxf(mx, rowp[c]);
        red[chunk * 16 + row] = mx;
        __syncthreads();
        if (tid < 16) {
            float m2 = red[tid];
            #pragma unroll
            for (int c = 1; c < 16; ++c) m2 = fmaxf(m2, red[c * 16 + tid]);
            rowmax[tid] = m2;
        }
        __syncthreads();
        const float rm = rowmax[row];
        float s = 0.f;
        #pragma unroll
        for (int c = 0; c < 16; ++c) {
            float p = __expf((rowp[c] - rm) * scale);
            rowp[c] = p;
            s += p;
        }
        red[chunk * 16 + row] = s;
        __syncthreads();
        if (tid < 16) {
            float t = 0.f;
            #pragma unroll
            for (int c = 0; c < 16; ++c) t += red[c * 16 + tid];
            rinv[tid] = 1.0f / t;
        }
        __syncthreads();
    }

    // ---------------- Pass 2: O = P V  (K-dim = 256, step 4; 6 d-tiles per wave) -------
    const float* vBase = seg + ll;         // + 2*token*768 + dtile col
    v8f zero = {};
    v8f oacc[6];
    #pragma unroll
    for (int j = 0; j < 6; ++j) oacc[j] = zero;

    const int dbase0 = (w * 6) * 16;
    for (int k0 = 0; k0 < LSEG; k0 += 4) {
        const int tok = k0 + 2 * hl;                     // token index for v=0 (v=1 -> tok+1)
        v2f a = *(const v2f*)(&Ssh[ll * SROW + tok]);    // A element (M=ll, K=tok+{0,1})
        const float* r0 = vBase + (size_t)(2 * tok) * D_MODEL;
        const float* r1 = r0 + 2 * D_MODEL;
        #pragma unroll
        for (int j = 0; j < 6; ++j) {
            const int dc = dbase0 + j * 16;
            v2f bf;
            bf.x = r0[dc];                               // B element (K=tok,   N=dc+ll)
            bf.y = r1[dc];                               // B element (K=tok+1, N=dc+ll)
            oacc[j] = __builtin_amdgcn_wmma_f32_16x16x4_f32(false, a, false, bf, (short)0, oacc[j], false, false);
        }
    }

    // store O tile, scaling each row by 1/rowsum
    float* oBase = O + ((size_t)b * (NSEG * LSEG) + (size_t)n * LSEG + (size_t)(qt * QT)) * D_MODEL + ll;
    #pragma unroll
    for (int j = 0; j < 6; ++j) {
        const int dc = dbase0 + j * 16;
        #pragma unroll
        for (int v = 0; v < 8; ++v) {
            const int m = v + 8 * hl;
            oBase[(size_t)m * D_MODEL + dc] = oacc[j][v] * rinv[m];
        }
    }
}

extern "C" void kernel_launch(void* const* d_in, const int* in_sizes, int n_in,
                              void* d_out, int out_size, void* d_ws, size_t ws_size,
                              hipStream_t stream) {
    (void)in_sizes; (void)n_in; (void)out_size; (void)d_ws; (void)ws_size;
    const float* x = (const float*)d_in[0];
    float* out = (float*)d_out;
    dim3 grid(BATCH * NSEG * (LSEG / QT));  // 4*32*16 = 2048 blocks
    dim3 block(256);                        // 8 wave32
    hipLaunchKernelGGL(dilated_attn_kernel, grid, block, 0, stream, x, out);
}